// GCDDLayer_11158325035388
// MI455X (gfx1250) — compile-verified
//
#include <hip/hip_runtime.h>
#include <cstdint>

// GCDD: 10 fused curvature-diffusion steps on (8,16,512,512) fp32.
// Roofline: ~2.8 GB total traffic -> ~120us floor @ 23.3 TB/s HBM; ping-pong
// buffers are largely L2-resident (128 MiB each vs 192 MB L2). ~50 VALU
// flops/elem/step keeps compute at the roofline crossover, so the kernel
// minimizes LDS transactions (register-blocked 4-wide/4x4 compute) and uses
// the CDNA5 data movers: TDM tensor_load_to_lds for interior tiles and
// global_load_async_to_lds_b64 for halos/border tiles.

namespace {
constexpr int W      = 512;
constexpr int TILE   = 64;
constexpr int UROWS  = TILE + 4;   // 68 rows incl. halo-2
constexpr int USTR   = 68;         // u-tile row stride == width (TDM layout)
constexpr int S1ROWS = 66;         // stage-1 (ux,uy) rows incl. halo-1
constexpr int S1STR  = 68;         // stage-1 stride (cols 66,67 are zero pad)
}

#if __has_builtin(__builtin_amdgcn_tensor_load_to_lds)
#define HAVE_TDM 1
#else
#define HAVE_TDM 0
#endif

__device__ __forceinline__ void async_ld_b64(unsigned lds_byte, unsigned goff,
                                             const float* sbase) {
  // vdst = LDS byte address, vaddr = 32-bit byte offset, saddr = 64-bit base.
  asm volatile("global_load_async_to_lds_b64 %0, %1, %2"
               :: "v"(lds_byte), "v"(goff), "s"(sbase) : "memory");
}

__device__ __forceinline__ void wait_async0() {
#if __has_builtin(__builtin_amdgcn_s_wait_asynccnt)
  __builtin_amdgcn_s_wait_asynccnt(0);
#else
  asm volatile("s_wait_asynccnt 0" ::: "memory");
#endif
}

__device__ __forceinline__ void wait_tensor0() {
#if __has_builtin(__builtin_amdgcn_s_wait_tensorcnt)
  __builtin_amdgcn_s_wait_tensorcnt(0);
#else
  asm volatile("s_wait_tensorcnt 0" ::: "memory");
#endif
}

#if HAVE_TDM
typedef unsigned int u32x4 __attribute__((ext_vector_type(4)));
typedef int          i32x4 __attribute__((ext_vector_type(4)));
typedef int          i32x8 __attribute__((ext_vector_type(8)));

// One TDM op: 68 rows x 64 cols of fp32 from a 512-stride plane into LDS,
// padding +4 dwords after every 64 dwords so rows land at stride 68 floats.
__device__ __forceinline__ void tdm_load_tile(const float* gtile, unsigned lds_byte) {
  const unsigned long long ga = (unsigned long long)(uintptr_t)gtile;
  u32x4 g0;
  g0[0] = 1u;                                    // count=1, user mode
  g0[1] = lds_byte;                              // lds_addr (bytes)
  g0[2] = (unsigned)(ga & 0xFFFFFFFFu);          // global_addr[31:0]
  g0[3] = (unsigned)((ga >> 32) & 0x01FFFFFFu)   // global_addr[56:32]
        | 0x80000000u;                           // type=2 (bits 127:126 = 10b)
  i32x8 g1;
  g1[0] = (int)(0x20000u        // data_size=2 (4B)
        | (1u << 20)            // pad_enable
        | (5u << 22)            // pad_interval: 64 dwords
        | (3u << 25));          // pad_amount: 4 dwords
  g1[1] = (int)((512u & 0xFFFFu) << 16);                      // tensor_dim0 lo16
  g1[2] = (int)(((512u >> 16) & 0xFFFFu) | ((512u & 0xFFFFu) << 16)); // dim0 hi | dim1 lo
  g1[3] = (int)(((512u >> 16) & 0xFFFFu) | ((unsigned)TILE << 16));   // dim1 hi | tile_dim0=64
  g1[4] = UROWS;                // tile_dim1=68 rows, tile_dim2=0 (unused)
  g1[5] = 512;                  // tensor_dim0_stride lo32
  g1[6] = 0;
  g1[7] = 0;
  const i32x4 gz = {0, 0, 0, 0};
#if __clang_major__ >= 23
  const i32x8 gz8 = {0, 0, 0, 0, 0, 0, 0, 0};
  __builtin_amdgcn_tensor_load_to_lds(g0, g1, gz, gz, gz8, 0);
#else
  __builtin_amdgcn_tensor_load_to_lds(g0, g1, gz, gz, 0);
#endif
}
#endif  // HAVE_TDM

__device__ __forceinline__ float clampf(float v, float lo, float hi) {
  return fminf(fmaxf(v, lo), hi);
}

// Load 6 consecutive floats (16B-aligned base quad + 8B-aligned tail pair).
__device__ __forceinline__ void load6(float d[6], const float* rowbase, int c) {
  const float4 a = *(const float4*)(rowbase + c);
  const float2 b = *(const float2*)(rowbase + c + 4);
  d[0] = a.x; d[1] = a.y; d[2] = a.z; d[3] = a.w; d[4] = b.x; d[5] = b.y;
}

__global__ __launch_bounds__(256)
void gcdd_step_kernel(const float* __restrict__ src,
                      float* __restrict__ dst,
                      const float* __restrict__ x0,
                      const float* __restrict__ alpha_p,
                      const float* __restrict__ beta_p,
                      int final_step)
{
  // +8 slack: stage-1's last column quad reads 2 floats past the tile end
  // (values are discarded by the in-domain predicate).
  __shared__ alignas(16) float uvt[UROWS * USTR + 8];  // u tile; reused for v
  __shared__ alignas(16) float s1x[S1ROWS * S1STR];    // unclipped ux
  __shared__ alignas(16) float s1y[S1ROWS * S1STR];    // unclipped uy

  const int tid = threadIdx.x;
  const int img = blockIdx.z;
  const int ty0 = blockIdx.y * TILE;
  const int tx0 = blockIdx.x * TILE;
  const size_t planeoff = (size_t)img * (W * W);
  const float* splane = src + planeoff;

  // ======== Phase 1: fill u tile (halo 2), zero outside the image =========
#if HAVE_TDM
  const bool y_interior = (ty0 != 0) && (ty0 != W - TILE);
  if (y_interior) {
    // TDM moves the 68x64 center block; wave 0 only (TDM ignores EXEC).
    if (tid < 32) {
      tdm_load_tile(splane + (size_t)(ty0 - 2) * W + tx0,
                    (unsigned)(uintptr_t)&uvt[2]);
    }
    // Async B64 for the 2+2 halo columns of each row (rows always in-image).
    for (int t = tid; t < UROWS * 2; t += 256) {
      const int i  = t >> 1;
      const int gy = ty0 - 2 + i;
      if (t & 1) {  // right halo: cols 66,67 <-> gx = tx0+64, tx0+65
        if (tx0 + TILE < W)
          async_ld_b64((unsigned)(uintptr_t)&uvt[i * USTR + 66],
                       (unsigned)((gy * W + tx0 + 64) * 4), splane);
        else
          *(float2*)&uvt[i * USTR + 66] = make_float2(0.f, 0.f);
      } else {      // left halo: cols 0,1 <-> gx = tx0-2, tx0-1
        if (tx0 > 0)
          async_ld_b64((unsigned)(uintptr_t)&uvt[i * USTR],
                       (unsigned)((gy * W + tx0 - 2) * 4), splane);
        else
          *(float2*)&uvt[i * USTR] = make_float2(0.f, 0.f);
      }
    }
  } else
#endif
  {
    // General path: 34 aligned B64 chunks per row, zero-fill outside image.
    for (int t = tid; t < UROWS * 34; t += 256) {
      const int i  = t / 34;
      const int ch = t - i * 34;
      const int gy = ty0 - 2 + i;
      const bool rowin = (unsigned)gy < (unsigned)W;
      if (ch == 0) {            // cols 0,1
        if (rowin && tx0 > 0)
          async_ld_b64((unsigned)(uintptr_t)&uvt[i * USTR],
                       (unsigned)((gy * W + tx0 - 2) * 4), splane);
        else
          *(float2*)&uvt[i * USTR] = make_float2(0.f, 0.f);
      } else if (ch == 33) {    // cols 66,67
        if (rowin && tx0 + TILE < W)
          async_ld_b64((unsigned)(uintptr_t)&uvt[i * USTR + 66],
                       (unsigned)((gy * W + tx0 + 64) * 4), splane);
        else
          *(float2*)&uvt[i * USTR + 66] = make_float2(0.f, 0.f);
      } else {                  // interior pair: cols 2+2k, always in-image x
        const int j = 2 * ch;   // ch=1..32 -> j=2..64
        if (rowin)
          async_ld_b64((unsigned)(uintptr_t)&uvt[i * USTR + j],
                       (unsigned)((gy * W + tx0 - 2 + j) * 4), splane);
        else
          *(float2*)&uvt[i * USTR + j] = make_float2(0.f, 0.f);
      }
    }
  }
  wait_async0();
  wait_tensor0();   // no-op for waves that issued no TDM
  __syncthreads();

  // ======== Phase 2: stage-1 Sobel (unclipped), 4 cells per iteration =====
  // s1 cell (r,c): center = u-tile (r+1,c+1); zero outside the image because
  // the reference zero-pads the *intermediate* before the second conv.
  for (int t = tid; t < S1ROWS * 17; t += 256) {
    const int r  = t / 17;
    const int c0 = (t - r * 17) << 2;     // 0..64 (cells 66,67 forced zero)
    const int gy = ty0 - 1 + r;
    float ox[4] = {0.f, 0.f, 0.f, 0.f};
    float oy[4] = {0.f, 0.f, 0.f, 0.f};
    if ((unsigned)gy < (unsigned)W) {
      float u0[6], u1[6], u2[6];
      load6(u0, &uvt[(r    ) * USTR], c0);
      load6(u1, &uvt[(r + 1) * USTR], c0);
      load6(u2, &uvt[(r + 2) * USTR], c0);
#pragma unroll
      for (int k = 0; k < 4; ++k) {
        const float vx = ((u0[k+2] - u0[k]) + 2.f * (u1[k+2] - u1[k])
                        + (u2[k+2] - u2[k])) * 0.125f;
        const float vy = ((u2[k] - u0[k]) + 2.f * (u2[k+1] - u0[k+1])
                        + (u2[k+2] - u0[k+2])) * 0.125f;
        const int  gx = tx0 - 1 + c0 + k;
        const bool in = ((unsigned)gx < (unsigned)W) && (c0 + k < S1ROWS);
        ox[k] = in ? vx : 0.f;
        oy[k] = in ? vy : 0.f;
      }
    }
    *(float4*)&s1x[r * S1STR + c0] = make_float4(ox[0], ox[1], ox[2], ox[3]);
    *(float4*)&s1y[r * S1STR + c0] = make_float4(oy[0], oy[1], oy[2], oy[3]);
  }
  __syncthreads();

  // ======== Phase 3: stage-2 Sobel + pointwise update, 4x4 per thread =====
  const float alpha = fabsf(alpha_p[0]);
  const float beta  = fabsf(beta_p[0]);
  {
    const int y = (tid >> 4) << 2;   // output block origin
    const int x = (tid & 15) << 2;
    float bx[6][6], by[6][6];
#pragma unroll
    for (int r = 0; r < 6; ++r) {
      load6(bx[r], &s1x[(y + r) * S1STR], x);
      load6(by[r], &s1y[(y + r) * S1STR], x);
    }
#pragma unroll
    for (int yy = 0; yy < 4; ++yy) {
      const float2 ua = *(const float2*)&uvt[(y + yy + 2) * USTR + (x + 2)];
      const float2 ub = *(const float2*)&uvt[(y + yy + 2) * USTR + (x + 4)];
      const float uc[4] = {ua.x, ua.y, ub.x, ub.y};
      float vrow[4];
#pragma unroll
      for (int k = 0; k < 4; ++k) {
        const float* r0x = bx[yy]; const float* r1x = bx[yy + 1]; const float* r2x = bx[yy + 2];
        const float* r0y = by[yy];                                const float* r2y = by[yy + 2];
        float uxx = ((r0x[k+2] - r0x[k]) + 2.f * (r1x[k+2] - r1x[k])
                   + (r2x[k+2] - r2x[k])) * 0.125f;
        float uxy = ((r2x[k] - r0x[k]) + 2.f * (r2x[k+1] - r0x[k+1])
                   + (r2x[k+2] - r0x[k+2])) * 0.125f;
        float uyy = ((r2y[k] - r0y[k]) + 2.f * (r2y[k+1] - r0y[k+1])
                   + (r2y[k+2] - r0y[k+2])) * 0.125f;
        const float ux = clampf(bx[yy + 1][k + 1], -10.f, 10.f);
        const float uy = clampf(by[yy + 1][k + 1], -10.f, 10.f);
        uxx = clampf(uxx, -10.f, 10.f);
        uxy = clampf(uxy, -10.f, 10.f);
        uyy = clampf(uyy, -10.f, 10.f);

        const float g2   = ux * ux + uy * uy;
        const float denK = fmaxf(1.0f + g2, 1e-6f);
        const float K    = clampf((uxx * uyy - uxy * uxy) / (denK * denK + 1e-6f),
                                  -5.f, 5.f);
        const float denH = sqrtf(1.0f + g2 + 1e-6f);
        float H = ((1.0f + uy * uy) * uxx - 2.0f * ux * uy * uxy
                 + (1.0f + ux * ux) * uyy) / (2.0f * denH * denH * denH + 1e-6f);
        H = clampf(H, -5.f, 5.f);
        const float diff = clampf(alpha * K + beta * H, -1.f, 1.f);
        vrow[k] = uc[k] + 0.01f * diff;
      }
      // v overwrites u at the same cells; only this thread touches them here.
      *(float2*)&uvt[(y + yy + 2) * USTR + (x + 2)] = make_float2(vrow[0], vrow[1]);
      *(float2*)&uvt[(y + yy + 2) * USTR + (x + 4)] = make_float2(vrow[2], vrow[3]);
    }
  }
  __syncthreads();

  // ======== Phase 4: edge-replicate halo refresh via clamped LDS reads; ===
  //          vectorized B128 global stores; fuse 0.7u+0.3x on last step.
  float* dplane = dst + planeoff;
  const float* xplane = x0 + planeoff;
  for (int t = tid; t < TILE * 16; t += 256) {
    const int y   = t >> 4;
    const int x   = (t & 15) << 2;
    const int gy  = ty0 + y;
    const int gx  = tx0 + x;
    const int lcy = min(max(gy, 1), W - 2) - ty0;   // clamp stays inside tile
    float v[4];
#pragma unroll
    for (int k = 0; k < 4; ++k) {
      const int lcx = min(max(gx + k, 1), W - 2) - tx0;
      v[k] = uvt[(lcy + 2) * USTR + (lcx + 2)];
    }
    float4 o = make_float4(v[0], v[1], v[2], v[3]);
    if (final_step) {
      const float4 xv = *(const float4*)&xplane[(size_t)gy * W + gx];
      o.x = 0.7f * o.x + 0.3f * xv.x;
      o.y = 0.7f * o.y + 0.3f * xv.y;
      o.z = 0.7f * o.z + 0.3f * xv.z;
      o.w = 0.7f * o.w + 0.3f * xv.w;
    }
    *(float4*)&dplane[(size_t)gy * W + gx] = o;
  }
}

extern "C" void kernel_launch(void* const* d_in, const int* in_sizes, int n_in,
                              void* d_out, int out_size, void* d_ws, size_t ws_size,
                              hipStream_t stream) {
  const float* x       = (const float*)d_in[0];
  const float* alpha_p = (const float*)d_in[1];
  const float* beta_p  = (const float*)d_in[2];
  float* out = (float*)d_out;
  float* ws  = (float*)d_ws;   // needs 8*16*512*512 floats = 128 MiB scratch

  dim3 grid(W / TILE, W / TILE, 8 * 16);
  dim3 block(256);

  const float* src = x;
  for (int s = 0; s < 10; ++s) {
    float* dst = (s % 2 == 0) ? ws : out;   // step 9 (last) lands in d_out
    gcdd_step_kernel<<<grid, block, 0, stream>>>(src, dst, x, alpha_p, beta_p,
                                                 (s == 9) ? 1 : 0);
    src = dst;
  }
}